// IslandLoss_30167850287416
// MI455X (gfx1250) — compile-verified
//
#include <hip/hip_runtime.h>
#include <math.h>

#define BATCH     65536
#define FEAT      512
#define NCLS      16
#define LAMDA     1.0f
#define LAMDA1    10.0f
#define SCALE_F   1.0f
#define EPSV      1e-9f

#define CHUNKS_PER_SAMPLE (FEAT / 4)           // 128 float4 per sample
#define TOTAL_CHUNKS      (BATCH * CHUNKS_PER_SAMPLE)
#define NBLOCKS_P1        1024
#define BLK               256

typedef float v2f __attribute__((ext_vector_type(2)));
typedef float v8f __attribute__((ext_vector_type(8)));

// ---------------------------------------------------------------------------
// Kernel 1: center loss partial sums.  HBM-bound: stream features with b128
// loads, centers staged once per block into LDS, deterministic block reduce.
// ---------------------------------------------------------------------------
__global__ __launch_bounds__(BLK) void island_center_partial(
    const float* __restrict__ x,      // [BATCH, FEAT]
    const int*   __restrict__ y,      // [BATCH]
    const float* __restrict__ c,      // [NCLS, FEAT]
    float*       __restrict__ partial)// [NBLOCKS_P1]
{
    __shared__ float4 cs[NCLS * CHUNKS_PER_SAMPLE];   // 32 KB of centers
    __shared__ float  red[BLK];

    const float4* c4 = (const float4*)c;
    for (int i = threadIdx.x; i < NCLS * CHUNKS_PER_SAMPLE; i += BLK)
        cs[i] = c4[i];
    __syncthreads();

    const float4* x4 = (const float4*)x;
    float acc = 0.0f;
    for (int idx = blockIdx.x * BLK + threadIdx.x; idx < TOTAL_CHUNKS;
         idx += NBLOCKS_P1 * BLK) {
        const int sample = idx >> 7;          // / CHUNKS_PER_SAMPLE
        const int off    = idx & (CHUNKS_PER_SAMPLE - 1);
        const int cls    = y[sample];
        const float4 xv  = x4[idx];
        const float4 cv  = cs[cls * CHUNKS_PER_SAMPLE + off];
        const float dx = xv.x - cv.x;
        const float dy = xv.y - cv.y;
        const float dz = xv.z - cv.z;
        const float dw = xv.w - cv.w;
        acc = fmaf(dx, dx, acc);
        acc = fmaf(dy, dy, acc);
        acc = fmaf(dz, dz, acc);
        acc = fmaf(dw, dw, acc);
    }

    red[threadIdx.x] = acc;
    __syncthreads();
    for (int s = BLK >> 1; s > 0; s >>= 1) {
        if (threadIdx.x < s) red[threadIdx.x] += red[threadIdx.x + s];
        __syncthreads();
    }
    if (threadIdx.x == 0) partial[blockIdx.x] = red[0];
}

// ---------------------------------------------------------------------------
// Kernel 2: single block.  Reduce partials; wave 0 computes gram = C*C^T via
// V_WMMA_F32_16X16X4_F32 (128 WMMA issues over K=512); norms come from the
// gram diagonal; 256 threads score the upper triangle; thread 0 combines.
// ---------------------------------------------------------------------------
__global__ __launch_bounds__(BLK) void island_finalize(
    const float* __restrict__ partial,   // [NBLOCKS_P1]
    const float* __restrict__ c,         // [NCLS, FEAT]
    float*       __restrict__ out)       // [1]
{
    __shared__ float red[BLK];
    __shared__ float gram_s[NCLS][NCLS + 1];
    __shared__ float norms[NCLS];

    // --- reduce center-loss partials (deterministic order) ---
    float acc = 0.0f;
    for (int i = threadIdx.x; i < NBLOCKS_P1; i += BLK) acc += partial[i];
    red[threadIdx.x] = acc;
    __syncthreads();
    for (int s = BLK >> 1; s > 0; s >>= 1) {
        if (threadIdx.x < s) red[threadIdx.x] += red[threadIdx.x + s];
        __syncthreads();
    }
    const float center_sum = red[0];
    __syncthreads();

    // --- wave 0: gram = C * C^T via f32 WMMA, K stepped by 4 ---
    if (threadIdx.x < 32) {
        const int lane = threadIdx.x;
        const int m    = lane & 15;              // A row == B col (symmetric)
        const int kg   = (lane >> 4) << 1;       // K-pair select: 0 or 2
        const float* __restrict__ row = c + m * FEAT;

        v8f accv = {};
        for (int k0 = 0; k0 < FEAT; k0 += 4) {
            v2f a;
            a.x = row[k0 + kg];
            a.y = row[k0 + kg + 1];
            // A (16x4 f32) and B (4x16 f32) hold identical per-lane data here.
            accv = __builtin_amdgcn_wmma_f32_16x16x4_f32(
                /*neg_a=*/false, a, /*neg_b=*/false, a,
                /*c_mod=*/(short)0, accv,
                /*reuse_a=*/false, /*reuse_b=*/false);
        }
        // C/D layout: VGPR j, lanes 0-15 -> M=j, lanes 16-31 -> M=j+8; N=lane%16
        const int mBase = (lane >> 4) << 3;
#pragma unroll
        for (int j = 0; j < 8; ++j)
            gram_s[mBase + j][m] = accv[j];
    }
    __syncthreads();

    if (threadIdx.x < NCLS)
        norms[threadIdx.x] = sqrtf(gram_s[threadIdx.x][threadIdx.x]);
    __syncthreads();

    // --- upper-triangle cosine sum: one (m,n) entry per thread ---
    const int m = threadIdx.x >> 4;
    const int n = threadIdx.x & 15;
    float isum = 0.0f;
    if (m < n) {
        const float denom = norms[m] * norms[n] + EPSV;
        isum = gram_s[m][n] / denom + 1.0f;
    }
    red[threadIdx.x] = isum;
    __syncthreads();
    for (int s = BLK >> 1; s > 0; s >>= 1) {
        if (threadIdx.x < s) red[threadIdx.x] += red[threadIdx.x + s];
        __syncthreads();
    }

    if (threadIdx.x == 0) {
        const float loss_center = 0.5f * center_sum * (SCALE_F / (float)BATCH);
        out[0] = LAMDA * (loss_center + LAMDA1 * red[0]);
    }
}

// ---------------------------------------------------------------------------
extern "C" void kernel_launch(void* const* d_in, const int* in_sizes, int n_in,
                              void* d_out, int out_size, void* d_ws, size_t ws_size,
                              hipStream_t stream) {
    const float* x = (const float*)d_in[0];   // output_features [B, D] f32
    const int*   y = (const int*)d_in[1];     // y_truth [B] int
    const float* c = (const float*)d_in[2];   // feature_centers [C, D] f32
    float* out     = (float*)d_out;
    float* partial = (float*)d_ws;            // NBLOCKS_P1 floats of scratch

    island_center_partial<<<NBLOCKS_P1, BLK, 0, stream>>>(x, y, c, partial);
    island_finalize<<<1, BLK, 0, stream>>>(partial, c, out);
}